// DetectionLoss_41334765256869
// MI455X (gfx1250) — compile-verified
//
#include <hip/hip_runtime.h>
#include <math.h>

typedef float v2f __attribute__((ext_vector_type(2)));
typedef float v8f __attribute__((ext_vector_type(8)));

#define NUM_CLASSES 30
#define BATCH 16
#define NBOX 64
#define CLS_W 1.0f
#define REG_W 5.0f
#define OBJ_W 1.0f

// ws float layout:
// [0..2]  softplus_sum per scale   (zero-initialized by det_pos_kernel)
// [3..5]  pos_obj_sum  per scale
// [6..8]  cls_ce_sum   per scale
// [9..11] reg_sl1_sum  per scale
// [12..14] npos        per scale (as float)

__global__ __launch_bounds__(1024)
void det_pos_kernel(const float* __restrict__ cls0, const float* __restrict__ reg0, const float* __restrict__ obj0,
                    const float* __restrict__ cls1, const float* __restrict__ reg1, const float* __restrict__ obj1,
                    const float* __restrict__ cls2, const float* __restrict__ reg2, const float* __restrict__ obj2,
                    const float* __restrict__ boxes, const int* __restrict__ labels,
                    float* __restrict__ ws)
{
    __shared__ int   s_flat[BATCH * NBOX];
    __shared__ int   s_lab[BATCH * NBOX];
    __shared__ float s_box[BATCH * NBOX * 4];
    __shared__ float s_acc[4];

    const int s = blockIdx.x;                 // scale id 0..2
    const float* cls_p = (s == 0) ? cls0 : (s == 1) ? cls1 : cls2;
    const float* reg_p = (s == 0) ? reg0 : (s == 1) ? reg1 : reg2;
    const float* obj_p = (s == 0) ? obj0 : (s == 1) ? obj1 : obj2;
    const int W = (s == 0) ? 80 : (s == 1) ? 40 : 20;
    const int H = W;
    const int HWp = H * W;

    const int tid = threadIdx.x;              // 0..1023 == b*64+n
    const int b = tid >> 6;
    const int n = tid & 63;

    // Async global->LDS staging (CDNA5): labels b32/lane, boxes b128/lane.
    // LDS dest offset = low 32 bits of the generic pointer to the shared array.
    {
        unsigned lab_off = (unsigned)(size_t)&s_lab[tid];
        unsigned box_off = (unsigned)(size_t)&s_box[tid * 4];
        const int*   glab = labels + tid;
        const float* gbox = boxes + tid * 4;
        asm volatile("global_load_async_to_lds_b32 %0, %1, off"
                     :: "v"(lab_off), "v"(glab) : "memory");
        asm volatile("global_load_async_to_lds_b128 %0, %1, off"
                     :: "v"(box_off), "v"(gbox) : "memory");
        asm volatile("s_wait_asynccnt 0" ::: "memory");
    }
    if (tid < 4) s_acc[tid] = 0.0f;
    __syncthreads();

    const float cx = s_box[tid * 4 + 0];
    const float cy = s_box[tid * 4 + 1];
    int gx = (int)(cx * (float)W); gx = min(max(gx, 0), W - 1);
    int gy = (int)(cy * (float)H); gy = min(max(gy, 0), H - 1);
    const int cell = gy * W + gx;             // cell within one image
    const int flat = b * HWp + cell;          // flat cell id (matches obj layout)

    s_flat[tid] = flat;
    __syncthreads();

    // Last-write-wins owner for this cell: highest n in same batch with same flat.
    bool owner = true;
    for (int n2 = n + 1; n2 < NBOX; ++n2)
        if (s_flat[(b << 6) + n2] == flat) { owner = false; break; }

    if (owner) {
        // cls target: every duplicate sets its label to 1; argmax tie -> lowest index.
        int ml = s_lab[tid];
        for (int n2 = 0; n2 < NBOX; ++n2)
            if (s_flat[(b << 6) + n2] == flat) ml = min(ml, s_lab[(b << 6) + n2]);

        // Cross-entropy at this cell (gather 30 strided floats).
        const float* crow = cls_p + (size_t)b * NUM_CLASSES * HWp + cell;
        float v[NUM_CLASSES];
        float m = -INFINITY;
        #pragma unroll
        for (int c = 0; c < NUM_CLASSES; ++c) { v[c] = crow[(size_t)c * HWp]; m = fmaxf(m, v[c]); }
        float se = 0.0f;
        #pragma unroll
        for (int c = 0; c < NUM_CLASSES; ++c) se += __expf(v[c] - m);
        const float ce = m + __logf(se) - v[ml];

        // Smooth-L1 (beta=1) vs the winning (this thread's) box, mean over 4, clamp 10.
        const float* rrow = reg_p + (size_t)b * 4 * HWp + cell;
        float sl1 = 0.0f;
        #pragma unroll
        for (int k = 0; k < 4; ++k) {
            const float d = fabsf(rrow[(size_t)k * HWp] - s_box[tid * 4 + k]);
            sl1 += (d < 1.0f) ? (0.5f * d * d) : (d - 0.5f);
        }
        sl1 *= 0.25f;
        sl1 = fminf(sl1, 10.0f);

        atomicAdd(&s_acc[0], ce);
        atomicAdd(&s_acc[1], sl1);
        atomicAdd(&s_acc[2], obj_p[flat]);   // sum of obj logits at positive cells
        atomicAdd(&s_acc[3], 1.0f);          // npos
    }
    __syncthreads();

    if (tid == 0) {
        ws[0  + s] = 0.0f;                   // init softplus accumulator for next kernel
        ws[3  + s] = s_acc[2];
        ws[6  + s] = s_acc[0];
        ws[9  + s] = s_acc[1];
        ws[12 + s] = s_acc[3];
    }
}

// Dense reduction: sum softplus(obj) over `count` elements (count % 64 == 0).
// WMMA f32 16x16x4 with all-ones A is the accumulator:
//   sum(all elements of D) grows by exactly 16 * sum(B-tile) per issue,
//   independent of the f32 B VGPR striping.
__global__ __launch_bounds__(256)
void det_obj_softplus_kernel(const float* __restrict__ obj, int count, float* __restrict__ accum)
{
    const int lane   = threadIdx.x & 31;
    const int wave   = (int)((blockIdx.x * blockDim.x + threadIdx.x) >> 5);
    const int nwaves = (int)((gridDim.x * blockDim.x) >> 5);
    const int ntiles = count >> 6;            // 64 elements per tile

    const float2* __restrict__ obj2 = (const float2*)obj;

    v2f a; a.x = 1.0f; a.y = 1.0f;            // all-ones A (16x4)
    v8f c = {};                               // 16x16 f32 accumulator

    for (int t = wave; t < ntiles; t += nwaves) {
        const int base2 = t << 5;             // index into float2 array (32 pairs/tile)
        // Speculative prefetch of this wave's next tile (silently dropped if OOB).
        __builtin_prefetch(obj2 + base2 + (nwaves << 5), 0, 0);
        const float2 x = obj2[base2 + lane];  // one coalesced b64 per lane
        v2f bd;
        bd.x = fmaxf(x.x, 0.0f) + log1pf(__expf(-fabsf(x.x)));   // stable softplus
        bd.y = fmaxf(x.y, 0.0f) + log1pf(__expf(-fabsf(x.y)));
        c = __builtin_amdgcn_wmma_f32_16x16x4_f32(false, a, false, bd, (short)0, c, false, false);
    }

    float p = c[0] + c[1] + c[2] + c[3] + c[4] + c[5] + c[6] + c[7];
    // wave32 XOR butterfly via ds_swizzle (bitmask mode: xor<<10 | and=0x1f)
    p += __int_as_float(__builtin_amdgcn_ds_swizzle(__float_as_int(p), 0x401f)); // xor 16
    p += __int_as_float(__builtin_amdgcn_ds_swizzle(__float_as_int(p), 0x201f)); // xor 8
    p += __int_as_float(__builtin_amdgcn_ds_swizzle(__float_as_int(p), 0x101f)); // xor 4
    p += __int_as_float(__builtin_amdgcn_ds_swizzle(__float_as_int(p), 0x081f)); // xor 2
    p += __int_as_float(__builtin_amdgcn_ds_swizzle(__float_as_int(p), 0x041f)); // xor 1
    if (lane == 0) atomicAdd(accum, p * (1.0f / 16.0f));
}

__global__ void det_finalize_kernel(const float* __restrict__ ws, float* __restrict__ out)
{
    const float Ms[3] = { 16.0f * 80 * 80, 16.0f * 40 * 40, 16.0f * 20 * 20 };
    float cls_s = 0.0f, reg_s = 0.0f, obj_s = 0.0f;
    for (int s = 0; s < 3; ++s) {
        const float npos = fmaxf(ws[12 + s], 1.0f);
        cls_s += ws[6 + s] / npos * CLS_W;
        reg_s += ws[9 + s] / npos * REG_W;
        obj_s += (ws[0 + s] - ws[3 + s]) / Ms[s] * OBJ_W;
    }
    cls_s *= (1.0f / 3.0f);
    reg_s *= (1.0f / 3.0f);
    obj_s *= (1.0f / 3.0f);
    out[0] = cls_s + reg_s + obj_s;
    out[1] = cls_s;
    out[2] = reg_s;
    out[3] = obj_s;
}

extern "C" void kernel_launch(void* const* d_in, const int* in_sizes, int n_in,
                              void* d_out, int out_size, void* d_ws, size_t ws_size,
                              hipStream_t stream)
{
    // setup_inputs() dict order: (cls,reg,obj) x 3 scales, then boxes, labels
    const float* cls0 = (const float*)d_in[0];
    const float* reg0 = (const float*)d_in[1];
    const float* obj0 = (const float*)d_in[2];
    const float* cls1 = (const float*)d_in[3];
    const float* reg1 = (const float*)d_in[4];
    const float* obj1 = (const float*)d_in[5];
    const float* cls2 = (const float*)d_in[6];
    const float* reg2 = (const float*)d_in[7];
    const float* obj2 = (const float*)d_in[8];
    const float* boxes  = (const float*)d_in[9];
    const int*   labels = (const int*)d_in[10];

    float* ws  = (float*)d_ws;
    float* out = (float*)d_out;

    // 1) positives: dedup assignments, CE + smooth-L1 + positive-obj sums; inits ws.
    det_pos_kernel<<<3, 1024, 0, stream>>>(cls0, reg0, obj0, cls1, reg1, obj1,
                                           cls2, reg2, obj2, boxes, labels, ws);

    // 2) dense softplus reductions (WMMA accumulate), one per scale.
    const float* objs[3]   = { obj0, obj1, obj2 };
    const int    counts[3] = { 16 * 80 * 80, 16 * 40 * 40, 16 * 20 * 20 };
    for (int s = 0; s < 3; ++s) {
        const int tiles = counts[s] >> 6;
        int blocks = (tiles + 7) / 8;          // 8 waves per 256-thread block
        if (blocks < 1)   blocks = 1;
        if (blocks > 256) blocks = 256;
        det_obj_softplus_kernel<<<blocks, 256, 0, stream>>>(objs[s], counts[s], ws + s);
    }

    // 3) combine.
    det_finalize_kernel<<<1, 1, 0, stream>>>(ws, out);
}